// ConvergedInhibition_66468913873360
// MI455X (gfx1250) — compile-verified
//
#include <hip/hip_runtime.h>

typedef __attribute__((ext_vector_type(2))) float v2f;
typedef __attribute__((ext_vector_type(8))) float v8f;

#define C_DIM     128
#define SCOPE_LEN 27
#define SPATIAL   4096   // 64 * 64
#define BATCH_N   64
#define W_STRIDE  132    // padded LDS row stride (132 % 64 == 4 -> conflict-free A reads)

// ---------------------------------------------------------------------------
// Kernel 1: build the 128x128 circulant deconvolution matrix W in workspace.
// out = ifft(fft(x)/Fk) along C  ==  circular conv with m = Re{ifft(1/Fk)},
// Fk = fft(delta - roll(pad(filter), C/2+1)).  Done in fp64 for accuracy.
// ---------------------------------------------------------------------------
__global__ __launch_bounds__(C_DIM) void build_w_kernel(
    const float* __restrict__ filt, float* __restrict__ Wg) {
  __shared__ double gsh[C_DIM];
  __shared__ double Fre[C_DIM];
  __shared__ double Fim[C_DIM];
  __shared__ float  msh[C_DIM];

  const int t = threadIdx.x;
  const double TWO_PI = 6.283185307179586476925286766559;

  // padded[i] = filt[i-50] for 50 <= i < 77 else 0   (pad_left = (128-27)/2 = 50)
  // rolled[i] = padded[(i - 65) mod 128]             (roll by C/2 + 1 = 65)
  // g[i]      = delta[i] - rolled[i]
  {
    int src = (t - 65) & (C_DIM - 1);
    double pad = (src >= 50 && src < 50 + SCOPE_LEN) ? (double)filt[src - 50] : 0.0;
    gsh[t] = ((t == 0) ? 1.0 : 0.0) - pad;
  }
  __syncthreads();

  // Fk[t] = sum_c g[c] * exp(-2*pi*i * t*c / 128)
  {
    double re = 0.0, im = 0.0;
    for (int c = 0; c < C_DIM; ++c) {
      double ang = -TWO_PI * (double)(t * c) / (double)C_DIM;
      re += gsh[c] * cos(ang);
      im += gsh[c] * sin(ang);
    }
    Fre[t] = re;
    Fim[t] = im;
  }
  __syncthreads();

  // m[t] = (1/128) * Re{ sum_f (1/Fk[f]) * exp(+2*pi*i * f*t / 128) }
  {
    double mre = 0.0;
    for (int f = 0; f < C_DIM; ++f) {
      double d  = Fre[f] * Fre[f] + Fim[f] * Fim[f];
      double ir =  Fre[f] / d;
      double ii = -Fim[f] / d;
      double ang = TWO_PI * (double)(f * t) / (double)C_DIM;
      mre += ir * cos(ang) - ii * sin(ang);
    }
    msh[t] = (float)(mre / (double)C_DIM);
  }
  __syncthreads();

  // W[i][j] = m[(i - j) mod 128]; thread t writes row t (row-major, stride 128)
  for (int j = 0; j < C_DIM; ++j) {
    Wg[t * C_DIM + j] = msh[(t - j) & (C_DIM - 1)];
  }
}

// ---------------------------------------------------------------------------
// Kernel 2: Out[n,:,s] = W @ Act[n,:,s] as a WMMA f32 GEMM.
// Block = 256 threads (8 waves). Each wave: 16 columns x all 128 rows.
// Grid.x = 64 batches * 32 column-tiles = 2048.
// ---------------------------------------------------------------------------
__global__ __launch_bounds__(256) void inhib_gemm_kernel(
    const float* __restrict__ act, const float* __restrict__ Wg,
    float* __restrict__ out) {
  __shared__ float Wlds[C_DIM * W_STRIDE];  // 67,584 B of the 320 KB WGP LDS

  const int t = threadIdx.x;

  // Stage W into LDS with padded row stride (coalesced global reads).
  for (int idx = t; idx < C_DIM * C_DIM; idx += 256) {
    int row = idx >> 7;
    int col = idx & (C_DIM - 1);
    Wlds[row * W_STRIDE + col] = Wg[idx];
  }
  __syncthreads();

  const int lane = t & 31;
  const int wave = t >> 5;
  const int half = lane >> 4;   // lane-half selects K pair {k0,k0+1} vs {k0+2,k0+3}
  const int lh   = lane & 15;

  const int n     = blockIdx.x >> 5;        // batch
  const int ctile = blockIdx.x & 31;        // 128-column tile within the 4096 spatial
  const int col   = ctile * 128 + wave * 16 + lh;

  const float* __restrict__ X = act + (size_t)n * C_DIM * SPATIAL + col;
  float* __restrict__       O = out + (size_t)n * C_DIM * SPATIAL + col;

  v8f acc[8];
#pragma unroll
  for (int r = 0; r < 8; ++r) {
    acc[r] = (v8f){0.f, 0.f, 0.f, 0.f, 0.f, 0.f, 0.f, 0.f};
  }

#pragma unroll 4
  for (int k0 = 0; k0 < C_DIM; k0 += 4) {
    const int krow = k0 + half * 2;

    // B fragment (4x16, K x N): lane-half holds rows {krow, krow+1} at column `col`.
    v2f b;
    b.x = X[(size_t)krow * SPATIAL];
    b.y = X[(size_t)(krow + 1) * SPATIAL];

#pragma unroll
    for (int r = 0; r < 8; ++r) {
      // A fragment (16x4, M x K) from LDS: row = r*16 + lh, cols {krow, krow+1}.
      // (row*132 + krow) is even -> 8-byte aligned ds_load_b64, conflict-free.
      v2f a = *(const v2f*)&Wlds[(r * 16 + lh) * W_STRIDE + krow];
      acc[r] = __builtin_amdgcn_wmma_f32_16x16x4_f32(
          false, a, false, b, (short)0, acc[r], false, false);
    }
  }

  // D layout (16x16 f32, 8 VGPRs): VGPR v -> M = v + 8*half, N = lh.
#pragma unroll
  for (int r = 0; r < 8; ++r) {
#pragma unroll
    for (int v = 0; v < 8; ++v) {
      const int m = r * 16 + v + half * 8;
      O[(size_t)m * SPATIAL] = acc[r][v];
    }
  }
}

// ---------------------------------------------------------------------------
// Harness entry point.
// ---------------------------------------------------------------------------
extern "C" void kernel_launch(void* const* d_in, const int* in_sizes, int n_in,
                              void* d_out, int out_size, void* d_ws, size_t ws_size,
                              hipStream_t stream) {
  const float* act  = (const float*)d_in[0];  // (64, 128, 64, 64) fp32
  const float* filt = (const float*)d_in[1];  // (1, 1, 27) fp32
  float* out = (float*)d_out;
  float* Wg  = (float*)d_ws;                  // 128*128 fp32 = 64 KB scratch

  build_w_kernel<<<1, C_DIM, 0, stream>>>(filt, Wg);
  inhib_gemm_kernel<<<BATCH_N * (SPATIAL / 128), 256, 0, stream>>>(act, Wg, out);
}